// RefLocal2_70085276336131
// MI455X (gfx1250) — compile-verified
//
#include <hip/hip_runtime.h>
#include <hip/hip_bf16.h>

// CDNA5 (gfx1250) local-attention kernel using V_WMMA_F32_16X16X4_F32.
// Memory-bound problem (~92MB min traffic @ 23.3TB/s ~= 4us), so we keep
// fp32 end-to-end (f32 WMMA) and focus on LDS tiling: each 8x8 query tile
// stages its 12x12 ref/ref_value region once (vs 25x re-read naively).
//
// v2: channel-PAIR-interleaved LDS layouts so every WMMA B-fragment is one
// ds_load_b64 straight into an even-aligned VGPR pair (no repack movs);
// XOR swizzle on the transposed ref store keeps staging conflicts <= 2-way
// while fragment reads stay conflict-free (halves on disjoint bank sets).
// v3: strength-reduce fragment addressing: (16t+nrow)^x == 16t + (nrow^x)
// for x<16, so all 9 B-fragment loads per k-step share ONE address register
// and differ only in the ds_load_b64 immediate offset (128*t bytes).

typedef float v2f __attribute__((ext_vector_type(2)));
typedef float v8f __attribute__((ext_vector_type(8)));

constexpr int B_ = 2, H_ = 128, W_ = 128, C_ = 256, V_ = 64;
constexpr int TILE = 8;            // 8x8 query tile per block
constexpr int RG   = TILE + 4;     // 12x12 region (k=5 -> halo 2)
constexpr int RPIX = RG * RG;      // 144 region pixels = 9 WMMA N-tiles
constexpr int NT   = RPIX / 16;    // 9
// s_refT: [C/2 pairs][144 swizzled pixels][2], pair stride 288 (== 32 mod 64)
constexpr int REFT_PS = 2 * RPIX;  // 288
// s_refv: [RPIX/2 pairs][80 v-slots][2], pair stride 160 (== 32 mod 64)
constexpr int REFV_PS = 160;
constexpr int P_PITCH = 164;       // s_p[row][r], 164%64=36, gcd(36,64)=4 -> 16-cycle
constexpr int LDS_FLOATS =
    (C_ / 2) * REFT_PS + (RPIX / 2) * REFV_PS + 64 * P_PITCH + 64;

__global__ __launch_bounds__(128) void local_attn_kernel(
    const float* __restrict__ mainq,  // [B,H,W,C]
    const float* __restrict__ mainv,  // [B,H,W,V]
    const float* __restrict__ refk,   // [B,H,W,C]
    const float* __restrict__ refv,   // [B,H,W,V]
    float* __restrict__ out)          // [B,H,W,V]
{
  extern __shared__ float smem[];
  float* s_refT = smem;                              // pair-interleaved, swizzled
  float* s_refv = s_refT + (C_ / 2) * REFT_PS;       // pair-interleaved
  float* s_p    = s_refv + (RPIX / 2) * REFV_PS;     // [64 rows][P_PITCH] exp-scores
  float* s_self = s_p + 64 * P_PITCH;                // [64] self scores

  const int b  = blockIdx.z;
  const int ty = blockIdx.y * TILE, tx = blockIdx.x * TILE;
  const int ry0 = ty - 2, rx0 = tx - 2;              // region origin (OOB -> zeros)
  const int tid = threadIdx.x;

  // ---------------- Phase 0: cooperative staging (128 threads) ---------------
  // ref region -> LDS, transposed + pair-interleaved:
  //   element(k, r) at (k>>1)*REFT_PS + 2*(r ^ ((k>>2)&15)) + (k&1)
  for (int idx = tid; idx < RPIX * (C_ / 4); idx += 128) {
    const int r  = idx >> 6;           // 64 float4 per region pixel
    const int c4 = idx & 63;
    const int gy = ry0 + r / RG, gx = rx0 + r % RG;
    float4 val = make_float4(0.f, 0.f, 0.f, 0.f);
    if ((unsigned)gy < (unsigned)H_ && (unsigned)gx < (unsigned)W_)
      val = *(const float4*)(refk + ((size_t)((b * H_ + gy) * W_ + gx)) * C_ + c4 * 4);
    const int c  = c4 * 4;
    const int rr = 2 * (r ^ ((c >> 2) & 15));        // swizzled pixel slot
    *(float2*)(s_refT + ((c >> 1) + 0) * REFT_PS + rr) = make_float2(val.x, val.y);
    *(float2*)(s_refT + ((c >> 1) + 1) * REFT_PS + rr) = make_float2(val.z, val.w);
  }
  // ref_value region -> LDS, pair-interleaved: element(k, v) at
  //   (k>>1)*REFV_PS + 2*v + (k&1)
  for (int idx = tid; idx < RPIX * (V_ / 4); idx += 128) {
    const int r  = idx >> 4;           // 16 float4 per region pixel
    const int v4 = idx & 15;
    const int gy = ry0 + r / RG, gx = rx0 + r % RG;
    float4 val = make_float4(0.f, 0.f, 0.f, 0.f);
    if ((unsigned)gy < (unsigned)H_ && (unsigned)gx < (unsigned)W_)
      val = *(const float4*)(refv + ((size_t)((b * H_ + gy) * W_ + gx)) * V_ + v4 * 4);
    const int base = (r >> 1) * REFV_PS + (r & 1);
    s_refv[base + 8 * v4 + 0] = val.x;
    s_refv[base + 8 * v4 + 2] = val.y;
    s_refv[base + 8 * v4 + 4] = val.z;
    s_refv[base + 8 * v4 + 6] = val.w;
  }
  __syncthreads();

  // ---------------- Per-wave setup: one 16-query M-tile per wave -------------
  const int w    = tid >> 5;           // wave 0..3 -> 4x4 query sub-block
  const int lane = tid & 31;
  const int nrow = lane & 15;          // M row (A) / N col (B) within WMMA tile
  const int khalf = (lane >> 4) * 2;   // f32 WMMA K pair: lanes 0-15 K{0,1}, 16-31 K{2,3}
  const int qy0 = (w >> 1) * 4, qx0 = (w & 1) * 4;

  // A-matrix row pointer (query = nrow of this M-tile), read from global/L0.
  const int aqy = qy0 + (nrow >> 2), aqx = qx0 + (nrow & 3);
  const float* aptr = mainq + ((size_t)((b * H_ + ty + aqy) * W_ + (tx + aqx))) * C_;

  // ---------------- GEMM1: S[16x144] = main[16x256] @ refT[256x144] ----------
  v8f acc[NT];
#pragma unroll
  for (int t = 0; t < NT; ++t) acc[t] = (v8f)(0.0f);

#pragma unroll 4
  for (int kc = 0; kc < C_ / 4; ++kc) {
    const int c = kc * 4;
    const v2f a = *(const v2f*)(aptr + c + khalf);
    const int x = (c >> 2) & 15;                     // swizzle const for this step
    // One shared fragment address per step; the 16t term becomes the
    // ds_load_b64 immediate offset (x<16 => (16t+nrow)^x == 16t + (nrow^x)).
    const float* frag = s_refT + ((c + khalf) >> 1) * REFT_PS + 2 * (nrow ^ x);
#pragma unroll
    for (int t = 0; t < NT; ++t) {
      const v2f bb = *(const v2f*)(frag + 32 * t);   // offset 128*t bytes
      acc[t] = __builtin_amdgcn_wmma_f32_16x16x4_f32(
          false, a, false, bb, (short)0, acc[t], false, false);
    }
  }

  // ---------------- Self score: dot(main,main) per query ---------------------
  {
    const float* sptr = aptr + (lane >> 4) * (C_ / 2);  // each half sums 128 ch
    float ssp = 0.f;
#pragma unroll 4
    for (int i = 0; i < C_ / 8; ++i) {
      float4 m4 = *(const float4*)(sptr + i * 4);
      ssp += m4.x * m4.x + m4.y * m4.y + m4.z * m4.z + m4.w * m4.w;
    }
    ssp += __shfl_xor(ssp, 16, 32);
    s_self[w * 16 + nrow] = ssp;
  }
  __syncthreads();  // uniform across all 4 waves (identical control flow)

  // ---------------- Mask to 5x5 windows + softmax ----------------------------
  // D layout: VGPR j holds row M=j (lanes 0-15) / M=j+8 (lanes 16-31), col=16t+nrow.
  const int Mb = (lane >> 4) * 8;
  int qyj[8], qxj[8];
#pragma unroll
  for (int j = 0; j < 8; ++j) {
    const int M = Mb + j;
    qyj[j] = qy0 + (M >> 2);
    qxj[j] = qx0 + (M & 3);
  }
  float rmax[8], rsum[8], esf[8], rden[8];
#pragma unroll
  for (int j = 0; j < 8; ++j) rmax[j] = -3.0e38f;
#pragma unroll
  for (int t = 0; t < NT; ++t) {
    const int r = t * 16 + nrow, ry = r / RG, rx = r % RG;
#pragma unroll
    for (int j = 0; j < 8; ++j) {
      const bool valid = ((unsigned)(ry - qyj[j]) <= 4u) && ((unsigned)(rx - qxj[j]) <= 4u);
      const float s = valid ? acc[t][j] : -3.0e38f;
      acc[t][j] = s;
      rmax[j] = fmaxf(rmax[j], s);
    }
  }
  // butterfly max across the 16 lanes holding each row (halves are independent)
#pragma unroll
  for (int j = 0; j < 8; ++j) {
    rmax[j] = fmaxf(rmax[j], __shfl_xor(rmax[j], 1, 32));
    rmax[j] = fmaxf(rmax[j], __shfl_xor(rmax[j], 2, 32));
    rmax[j] = fmaxf(rmax[j], __shfl_xor(rmax[j], 4, 32));
    rmax[j] = fmaxf(rmax[j], __shfl_xor(rmax[j], 8, 32));
  }
#pragma unroll
  for (int j = 0; j < 8; ++j) {
    const float ss = s_self[w * 16 + Mb + j];
    const float tm = fmaxf(rmax[j], ss);
    rmax[j] = tm;
    esf[j]  = __expf(ss - tm);
    rsum[j] = 0.f;
  }
#pragma unroll
  for (int t = 0; t < NT; ++t) {
#pragma unroll
    for (int j = 0; j < 8; ++j) {
      const float e = __expf(acc[t][j] - rmax[j]);  // masked -> exp(-huge)=0
      acc[t][j] = e;
      rsum[j] += e;
    }
  }
#pragma unroll
  for (int j = 0; j < 8; ++j) {
    rsum[j] += __shfl_xor(rsum[j], 1, 32);
    rsum[j] += __shfl_xor(rsum[j], 2, 32);
    rsum[j] += __shfl_xor(rsum[j], 4, 32);
    rsum[j] += __shfl_xor(rsum[j], 8, 32);
    rden[j] = 1.0f / (rsum[j] + esf[j]);
  }
  // Spill exp-scores P to LDS in A-matrix-friendly row-major form.
#pragma unroll
  for (int t = 0; t < NT; ++t) {
#pragma unroll
    for (int j = 0; j < 8; ++j)
      s_p[((size_t)(w * 16 + Mb + j)) * P_PITCH + t * 16 + nrow] = acc[t][j];
  }

  // ---------------- GEMM2: out[16x64] = P[16x144] @ refv[144x64] -------------
  v8f oacc[4];
#pragma unroll
  for (int vt = 0; vt < 4; ++vt) oacc[vt] = (v8f)(0.0f);
  const float* prow = s_p + ((size_t)(w * 16 + nrow)) * P_PITCH;
#pragma unroll 2
  for (int kr = 0; kr < RPIX / 4; ++kr) {
    const int k = kr * 4;
    const v2f a = *(const v2f*)(prow + k + khalf);
    const float* frag = s_refv + ((k + khalf) >> 1) * REFV_PS + 2 * nrow;
#pragma unroll
    for (int vt = 0; vt < 4; ++vt) {
      const v2f bb = *(const v2f*)(frag + 32 * vt);  // offset 128*vt bytes
      oacc[vt] = __builtin_amdgcn_wmma_f32_16x16x4_f32(
          false, a, false, bb, (short)0, oacc[vt], false, false);
    }
  }

  // ---------------- Epilogue: add self term, normalize, store ----------------
#pragma unroll
  for (int j = 0; j < 8; ++j) {
    const size_t base = ((size_t)((b * H_ + ty + qyj[j]) * W_ + (tx + qxj[j]))) * V_;
#pragma unroll
    for (int vt = 0; vt < 4; ++vt) {
      const int v = vt * 16 + nrow;          // 16 lanes -> 64B coalesced
      const float mv = mainv[base + v];
      out[base + v] = (oacc[vt][j] + esf[j] * mv) * rden[j];
    }
  }
}

extern "C" void kernel_launch(void* const* d_in, const int* in_sizes, int n_in,
                              void* d_out, int out_size, void* d_ws, size_t ws_size,
                              hipStream_t stream) {
  (void)in_sizes; (void)n_in; (void)d_ws; (void)ws_size; (void)out_size;
  const float* mainq = (const float*)d_in[0];
  const float* mainv = (const float*)d_in[1];
  const float* refk  = (const float*)d_in[2];
  const float* refv  = (const float*)d_in[3];
  float* outp = (float*)d_out;

  constexpr size_t shmem = (size_t)LDS_FLOATS * sizeof(float);  // 235,776 B < 320 KB
  static_assert(shmem <= 320 * 1024, "LDS budget exceeded");
  (void)hipFuncSetAttribute((const void*)local_attn_kernel,
                            hipFuncAttributeMaxDynamicSharedMemorySize, (int)shmem);

  dim3 grid(W_ / TILE, H_ / TILE, B_);
  dim3 block(128);
  local_attn_kernel<<<grid, block, shmem, stream>>>(mainq, mainv, refk, refv, outp);
}